// ConditionalEncoder_66142496358642
// MI455X (gfx1250) — compile-verified
//
#include <hip/hip_runtime.h>

typedef float v2f __attribute__((ext_vector_type(2)));
typedef float v8f __attribute__((ext_vector_type(8)));

#define B_ 16
#define H_ 192
#define S_ 1024
#define P_ 128
#define T_ 8192

// ---------------------------------------------------------------------------
// Kernel 1: token-level GEMM   E[b,o,s] = sum_h W_enh[o,h] * emb[b,h,s]
// One wave32 computes a 16(M=o) x 16(N=s) f32 tile with V_WMMA_F32_16X16X4_F32,
// stepping K = H in chunks of 4.
//
// VGPR layouts per CDNA5 ISA 7.12.2 (32-bit data):
//   A 16x4 : VGPR j, lane-half g -> element [M = lane%16][K = 2*g + j]
//   B 4x16 : VGPR j, lane-half g -> element [K = 2*g + j][N = lane%16]
//   C/D    : VGPR r, lane-half g -> element [M = r + 8*g][N = lane%16]
// ---------------------------------------------------------------------------
__global__ __launch_bounds__(256) void gemm_token_kernel(
    const float* __restrict__ W,    // [H_,H_] row-major (o,h)
    const float* __restrict__ emb,  // [B_,H_,S_]
    float* __restrict__ E)          // [B_,H_,S_]
{
    const int lane = threadIdx.x & 31;
    const int wave = threadIdx.x >> 5;
    const int id   = blockIdx.x * 8 + wave;

    const int tilesPerB = (H_ / 16) * (S_ / 16);   // 12 * 64 = 768
    const int b  = id / tilesPerB;
    const int r  = id % tilesPerB;
    const int mt = r / (S_ / 16);
    const int nt = r % (S_ / 16);

    const float* Xb = emb + (size_t)b * H_ * S_;
    const int g    = lane >> 4;          // lane half: 0 or 1
    const int l16  = lane & 15;
    const int mrow = mt * 16 + l16;      // A-matrix M for this lane
    const int ncol = nt * 16 + l16;      // B/C-matrix N for this lane

    v8f acc = {};
    for (int k = 0; k < H_; k += 4) {
        v2f a;
        a.x = W[(size_t)mrow * H_ + k + 2 * g + 0];
        a.y = W[(size_t)mrow * H_ + k + 2 * g + 1];
        v2f bm;
        bm.x = Xb[(size_t)(k + 2 * g + 0) * S_ + ncol];
        bm.y = Xb[(size_t)(k + 2 * g + 1) * S_ + ncol];
        // 8 args: (neg_a, A, neg_b, B, c_mod, C, reuse_a, reuse_b)
        acc = __builtin_amdgcn_wmma_f32_16x16x4_f32(
            false, a, false, bm, (short)0, acc, false, false);
    }

    float* Eb = E + (size_t)b * H_ * S_;
    #pragma unroll
    for (int rr = 0; rr < 8; ++rr)
        Eb[(size_t)(mt * 16 + rr + 8 * g) * S_ + ncol] = acc[rr];
}

// ---------------------------------------------------------------------------
// Kernel 2: per-batch inclusive scan of durations (S_=1024, one block/batch)
// ---------------------------------------------------------------------------
__global__ __launch_bounds__(1024) void scan_kernel(
    const int* __restrict__ dur, int* __restrict__ csum)
{
    __shared__ int sm[S_];
    const int b = blockIdx.x, t = threadIdx.x;
    sm[t] = dur[b * S_ + t];
    __syncthreads();
    for (int off = 1; off < S_; off <<= 1) {
        int v = (t >= off) ? sm[t - off] : 0;
        __syncthreads();
        sm[t] += v;
        __syncthreads();
    }
    csum[b * S_ + t] = sm[t];
}

// ---------------------------------------------------------------------------
// Kernel 3: frame expansion (bandwidth-bound bulk: 100.7 MB of writes)
// grid = B_ * (T_/256); each thread owns one frame t, binary-searches the
// LDS-resident cumsum, then streams 192 coalesced column writes.
// ---------------------------------------------------------------------------
__global__ __launch_bounds__(256) void expand_kernel(
    const float* __restrict__ E, const int* __restrict__ csum,
    float* __restrict__ out)
{
    __shared__ int cs[S_];
    const int b     = blockIdx.x >> 5;          // T_/256 = 32 frame-blocks / b
    const int tbase = (blockIdx.x & 31) * 256;
    for (int i = threadIdx.x; i < S_; i += 256) cs[i] = csum[b * S_ + i];
    __syncthreads();

    const int t     = tbase + (int)threadIdx.x;
    const int total = cs[S_ - 1];
    // searchsorted(csum, t, side='right')
    int lo = 0, hi = S_;
    while (lo < hi) {
        int mid = (lo + hi) >> 1;
        if (cs[mid] <= t) lo = mid + 1; else hi = mid;
    }
    const int idx   = (lo < S_) ? lo : (S_ - 1);
    const bool live = (t < total);

    const float* Eb = E + (size_t)b * H_ * S_;
    float* ob = out + (size_t)b * H_ * T_ + t;
    #pragma unroll 4
    for (int o = 0; o < H_; ++o)
        ob[(size_t)o * T_] = live ? Eb[(size_t)o * S_ + idx] : 0.0f;
}

// ---------------------------------------------------------------------------
// Kernel 4: prior[b,p] = b_prior[p] + sum_o W_prior[o,p] * mean_enh[b,o]
// where mean_enh[b,o] = (1/T_) * sum_s E[b,o,s] * dur[b,s]   (duration-weighted
// token sum == mean over frames, since masked frames contribute zero).
// ---------------------------------------------------------------------------
__global__ __launch_bounds__(256) void prior_kernel(
    const float* __restrict__ E, const int* __restrict__ dur,
    const float* __restrict__ Wp, const float* __restrict__ bp,
    float* __restrict__ prior)
{
    __shared__ float df[S_];
    __shared__ float menh[H_];
    const int b = blockIdx.x, tid = threadIdx.x;
    for (int i = tid; i < S_; i += 256) df[i] = (float)dur[b * S_ + i];
    __syncthreads();
    if (tid < H_) {
        const float* Er = E + (size_t)b * H_ * S_ + (size_t)tid * S_;
        float s = 0.0f;
        for (int j = 0; j < S_; ++j) s += Er[j] * df[j];
        menh[tid] = s * (1.0f / (float)T_);
    }
    __syncthreads();
    if (tid < P_) {
        float p = bp[tid];
        for (int o = 0; o < H_; ++o) p += menh[o] * Wp[(size_t)o * P_ + tid];
        prior[(size_t)b * P_ + tid] = p;
    }
}

// ---------------------------------------------------------------------------
extern "C" void kernel_launch(void* const* d_in, const int* in_sizes, int n_in,
                              void* d_out, int out_size, void* d_ws, size_t ws_size,
                              hipStream_t stream) {
    const float* emb = (const float*)d_in[0];   // [B,H,S] f32
    const int*   dur = (const int*)d_in[1];     // [B,S] i32
    const float* W   = (const float*)d_in[2];   // [H,H] f32
    const float* Wp  = (const float*)d_in[3];   // [H,P] f32
    const float* bp  = (const float*)d_in[4];   // [P] f32

    float* out = (float*)d_out;                 // enhanced [B,H,T] then prior [B,P]
    float* E   = (float*)d_ws;                                    // B*H*S f32
    int* csum  = (int*)((char*)d_ws + (size_t)B_ * H_ * S_ * sizeof(float));

    // 1) token-level GEMM via f32 WMMA: B_*768 tiles, 8 waves/block
    gemm_token_kernel<<<(B_ * (H_ / 16) * (S_ / 16)) / 8, 256, 0, stream>>>(W, emb, E);
    // 2) per-batch cumsum of durations
    scan_kernel<<<B_, 1024, 0, stream>>>(dur, csum);
    // 3) expansion (dominant HBM write stream)
    expand_kernel<<<B_ * (T_ / 256), 256, 0, stream>>>(E, csum, out);
    // 4) prior head
    prior_kernel<<<B_, 256, 0, stream>>>(E, dur, Wp, bp, out + (size_t)B_ * H_ * T_);
}